// CrossGGInversed_88639535055061
// MI455X (gfx1250) — compile-verified
//
#include <hip/hip_runtime.h>
#include <hip/hip_bf16.h>

// GAT layer: N nodes, E edges, F=128 in-feats, H=8 heads, D=16 head-dim.
// Phase 1: Wh = h @ W  via V_WMMA_F32_16X16X4_F32 (full f32 precision).
// Phase 2: factored attention logits + segment softmax + weighted scatter.

static constexpr int KF  = 128;  // input features
static constexpr int KH  = 8;    // heads
static constexpr int KD  = 16;   // head dim
static constexpr int KHD = 128;  // H*D
static constexpr float KALPHA = 0.2f;

typedef __attribute__((ext_vector_type(2))) float v2f;
typedef __attribute__((ext_vector_type(8))) float v8f;

// ---------- ordered-uint encoding for float atomic max ----------
__device__ __forceinline__ unsigned fmax_enc(float f) {
  unsigned u = __float_as_uint(f);
  return (u & 0x80000000u) ? ~u : (u | 0x80000000u);
}
__device__ __forceinline__ float fmax_dec(unsigned u) {
  return (u & 0x80000000u) ? __uint_as_float(u & 0x7fffffffu)
                           : __uint_as_float(~u);
}
// enc(-inf) = 0x007FFFFF
static constexpr unsigned ENC_NEG_INF = 0x007FFFFFu;

// ---------- init kernels (out/z/m are poisoned by harness; re-init per call) ----------
__global__ void init_nh_kernel(float* __restrict__ z, unsigned* __restrict__ menc, int count) {
  int i = blockIdx.x * blockDim.x + threadIdx.x;
  if (i < count) { z[i] = 0.0f; menc[i] = ENC_NEG_INF; }
}
__global__ void zero_kernel(float* __restrict__ p, int count) {
  int i = blockIdx.x * blockDim.x + threadIdx.x;
  if (i < count) p[i] = 0.0f;
}

// ---------- Phase 1: Wh[n, h*16+d] = sum_f h[n,f] * W[h,f,d] ----------
// One 256-thread block = 8 waves; block computes a 16-row strip of Wh.
// Wave w computes the 16x16 tile for head w (columns 16w..16w+15), K=128
// via 32 chained V_WMMA_F32_16X16X4_F32.
static constexpr int AS_STRIDE = 132;  // pad 128->132 floats: bank-conflict-free A reads

__global__ __launch_bounds__(256) void gat_gemm_wmma_kernel(
    const float* __restrict__ hfeat, const float* __restrict__ W,
    float* __restrict__ Wh, int N) {
  __shared__ float As[16 * AS_STRIDE];  // 16 rows of h, padded

  const int tid  = threadIdx.x;
  const int row0 = blockIdx.x * 16;

  // Stage the 16x128 strip of h (coalesced), zero-fill past N.
  for (int i = tid; i < 16 * KF; i += 256) {
    int r = i >> 7, f = i & (KF - 1);
    float v = 0.0f;
    if (row0 + r < N) v = hfeat[(size_t)(row0 + r) * KF + f];
    As[r * AS_STRIDE + f] = v;
  }
  __syncthreads();

  const int wave = tid >> 5;        // head / column tile
  const int lane = tid & 31;
  const int hl   = lane >> 4;       // half: 0 => K+0,K+1 ; 1 => K+2,K+3
  const int col  = lane & 15;       // A: row M ; B: column N ; D: column N
  const int n0   = wave * 16;

  // B fragment source: B[f][n0+col] = W[wave*2048 + f*16 + col] (W is [H,F,D]).
  const float* Wb = W + wave * (KF * KD) + col;

  v8f acc = {};
  for (int k0 = 0; k0 < KF; k0 += 4) {
    const int k = k0 + 2 * hl;
    v2f a, b;
    a.x = As[col * AS_STRIDE + k];
    a.y = As[col * AS_STRIDE + k + 1];
    b.x = Wb[(size_t)k * KD];
    b.y = Wb[(size_t)(k + 1) * KD];
    acc = __builtin_amdgcn_wmma_f32_16x16x4_f32(
        false, a, false, b, (short)0, acc, false, false);
  }

  // D layout: VGPR v -> row M = v + 8*hl, column N = col.
#pragma unroll
  for (int v = 0; v < 8; ++v) {
    int r = row0 + v + 8 * hl;
    if (r < N) Wh[(size_t)r * KHD + n0 + col] = acc[v];
  }
}

// ---------- Phase 2a: per-node scores s1,s2 ----------
__global__ void scores_kernel(const float* __restrict__ Wh,
                              const float* __restrict__ a1,
                              const float* __restrict__ a2,
                              float* __restrict__ s1, float* __restrict__ s2,
                              int N) {
  int idx = blockIdx.x * blockDim.x + threadIdx.x;
  if (idx >= N * KH) return;
  int n = idx >> 3, hh = idx & 7;
  const float* w  = Wh + (size_t)n * KHD + hh * KD;
  const float* v1 = a1 + hh * KD;
  const float* v2 = a2 + hh * KD;
  float acc1 = 0.0f, acc2 = 0.0f;
#pragma unroll
  for (int d = 0; d < KD; ++d) {
    float v = w[d];
    acc1 += v * v1[d];
    acc2 += v * v2[d];
  }
  s1[idx] = acc1;
  s2[idx] = acc2;
}

__device__ __forceinline__ float edge_logit(const float* s1, const float* s2,
                                            int s, int d, int hh) {
  float e = s1[s * KH + hh] + s2[d * KH + hh];
  return (e > 0.0f) ? e : KALPHA * e;  // leaky relu
}

// ---------- Phase 2b: segment max over dst ----------
__global__ void edge_max_kernel(const int* __restrict__ src, const int* __restrict__ dst,
                                const float* __restrict__ s1, const float* __restrict__ s2,
                                unsigned* __restrict__ menc, int E) {
  int idx = blockIdx.x * blockDim.x + threadIdx.x;
  if (idx >= E * KH) return;
  int e = idx >> 3, hh = idx & 7;
  int d = dst[e];
  float lv = edge_logit(s1, s2, src[e], d, hh);
  atomicMax(&menc[d * KH + hh], fmax_enc(lv));
}

__global__ void finalize_max_kernel(const unsigned* __restrict__ menc,
                                    float* __restrict__ m, int count) {
  int i = blockIdx.x * blockDim.x + threadIdx.x;
  if (i >= count) return;
  float v = fmax_dec(menc[i]);
  m[i] = (v < -3.0e38f) ? 0.0f : v;  // empty segments (-inf) -> 0
}

// ---------- Phase 2c: exp-sum over dst ----------
__global__ void edge_expsum_kernel(const int* __restrict__ src, const int* __restrict__ dst,
                                   const float* __restrict__ s1, const float* __restrict__ s2,
                                   const float* __restrict__ m, float* __restrict__ z, int E) {
  int idx = blockIdx.x * blockDim.x + threadIdx.x;
  if (idx >= E * KH) return;
  int e = idx >> 3, hh = idx & 7;
  int d = dst[e];
  float lv = edge_logit(s1, s2, src[e], d, hh);
  float p = __expf(lv - m[d * KH + hh]);
  unsafeAtomicAdd(&z[d * KH + hh], p);
}

__global__ void finalize_z_kernel(float* __restrict__ z, int count) {
  int i = blockIdx.x * blockDim.x + threadIdx.x;
  if (i >= count) return;
  z[i] = 1.0f / fmaxf(z[i], 1e-16f);  // in place -> zinv
}

// ---------- Phase 2d: weighted scatter of messages ----------
__global__ void edge_scatter_kernel(const int* __restrict__ src, const int* __restrict__ dst,
                                    const float* __restrict__ s1, const float* __restrict__ s2,
                                    const float* __restrict__ m, const float* __restrict__ zinv,
                                    const float* __restrict__ Wh, float* __restrict__ out, int E) {
  int idx = blockIdx.x * blockDim.x + threadIdx.x;
  if (idx >= E * KH) return;
  int e = idx >> 3, hh = idx & 7;
  int s = src[e], d = dst[e];
  float lv   = edge_logit(s1, s2, s, d, hh);
  float attn = __expf(lv - m[d * KH + hh]) * zinv[d * KH + hh];
  const float4* wsrc = (const float4*)(Wh + (size_t)s * KHD + hh * KD);
  float* op = out + (size_t)d * KHD + hh * KD;
#pragma unroll
  for (int q = 0; q < 4; ++q) {
    float4 w4 = wsrc[q];
    unsafeAtomicAdd(op + q * 4 + 0, attn * w4.x);
    unsafeAtomicAdd(op + q * 4 + 1, attn * w4.y);
    unsafeAtomicAdd(op + q * 4 + 2, attn * w4.z);
    unsafeAtomicAdd(op + q * 4 + 3, attn * w4.w);
  }
}

extern "C" void kernel_launch(void* const* d_in, const int* in_sizes, int n_in,
                              void* d_out, int out_size, void* d_ws, size_t ws_size,
                              hipStream_t stream) {
  const float* hfeat = (const float*)d_in[0];   // [N, 128]
  const int*   src   = (const int*)d_in[1];     // [E]
  const int*   dst   = (const int*)d_in[2];     // [E]
  const float* W     = (const float*)d_in[3];   // [8, 128, 16]
  const float* a1    = (const float*)d_in[4];   // [8, 16]
  const float* a2    = (const float*)d_in[5];   // [8, 16]
  float* out = (float*)d_out;                   // [N, 128]

  const int N = in_sizes[0] / KF;
  const int E = in_sizes[1];

  // Workspace layout (floats): Wh[N*128] | s1[N*8] | s2[N*8] | menc[N*8] | m[N*8] | z[N*8]
  float*    Wh   = (float*)d_ws;
  float*    s1   = Wh + (size_t)N * KHD;
  float*    s2   = s1 + (size_t)N * KH;
  unsigned* menc = (unsigned*)(s2 + (size_t)N * KH);
  float*    m    = (float*)(menc + (size_t)N * KH);
  float*    z    = m + (size_t)N * KH;

  const int TB = 256;
  const int nh      = N * KH;
  const int nhd     = N * KHD;
  const int eh      = E * KH;
  const int g_nh    = (nh  + TB - 1) / TB;
  const int g_nhd   = (nhd + TB - 1) / TB;
  const int g_eh    = (eh  + TB - 1) / TB;
  const int g_gemm  = (N + 15) / 16;

  // init (out/z poisoned each timing run -> must clear every call)
  init_nh_kernel<<<g_nh, TB, 0, stream>>>(z, menc, nh);
  zero_kernel<<<g_nhd, TB, 0, stream>>>(out, nhd);

  // Phase 1: projection GEMM (WMMA f32)
  gat_gemm_wmma_kernel<<<g_gemm, TB, 0, stream>>>(hfeat, W, Wh, N);

  // Phase 2: scores -> segment max -> exp-sum -> normalize -> scatter
  scores_kernel<<<g_nh, TB, 0, stream>>>(Wh, a1, a2, s1, s2, N);
  edge_max_kernel<<<g_eh, TB, 0, stream>>>(src, dst, s1, s2, menc, E);
  finalize_max_kernel<<<g_nh, TB, 0, stream>>>(menc, m, nh);
  edge_expsum_kernel<<<g_eh, TB, 0, stream>>>(src, dst, s1, s2, m, z, E);
  finalize_z_kernel<<<g_nh, TB, 0, stream>>>(z, nh);
  edge_scatter_kernel<<<g_eh, TB, 0, stream>>>(src, dst, s1, s2, m, z, Wh, out, E);
}